// PhaseCell_87230785782059
// MI455X (gfx1250) — compile-verified
//
#include <hip/hip_runtime.h>
#include <hip/hip_bf16.h>

// ---------------------------------------------------------------------------
// PhaseCell lookup-table kernel for MI455X (gfx1250).
//
// Memory-bound: 36 B of HBM traffic per element (16 B int32 idx in, 20 B f32
// out) * 8.39M elems ~= 302 MB  => ~13 us floor at 23.3 TB/s. Tables (16 KB)
// live in LDS. Final "strength" reduction uses a chained
// V_WMMA_F32_16X16X4_F32 with B = ones (full f32 precision, deterministic).
// ---------------------------------------------------------------------------

typedef __attribute__((ext_vector_type(2))) float v2f;
typedef __attribute__((ext_vector_type(8))) float v8f;

#define TBL 1024
#define MOD_MASK 1023

__global__ __launch_bounds__(256) void phasecell_main_kernel(
    const int* __restrict__ ctx_p,  const int* __restrict__ ctx_m,
    const int* __restrict__ self_p, const int* __restrict__ self_m,
    const float* __restrict__ cos_t,  const float* __restrict__ exp_t,
    const float* __restrict__ dcos_t, const float* __restrict__ dexp_t,
    float* __restrict__ phase_out, float* __restrict__ mag_out,
    float* __restrict__ signal_out, float* __restrict__ gph_out,
    float* __restrict__ gmg_out, float* __restrict__ partials,
    long long nvec, long long D)
{
    __shared__ float s_cos[TBL];
    __shared__ float s_exp[TBL];
    __shared__ float s_dcos[TBL];
    __shared__ float s_dexp[TBL];
    __shared__ float s_wsum[8];

    for (int i = threadIdx.x; i < TBL; i += blockDim.x) {
        s_cos[i]  = cos_t[i];
        s_exp[i]  = exp_t[i];
        s_dcos[i] = dcos_t[i];
        s_dexp[i] = dexp_t[i];
    }
    __syncthreads();

    const int4* __restrict__ cp4 = (const int4*)ctx_p;
    const int4* __restrict__ cm4 = (const int4*)ctx_m;
    const int4* __restrict__ sp4 = (const int4*)self_p;
    const int4* __restrict__ sm4 = (const int4*)self_m;

    float acc = 0.0f;
    const long long stride = (long long)gridDim.x * blockDim.x;

    for (long long v = (long long)blockIdx.x * blockDim.x + threadIdx.x;
         v < nvec; v += stride) {
        int4 cp = cp4[v];
        int4 sp = sp4[v];
        int4 cm = cm4[v];
        int4 sm = sm4[v];

        int p0 = (cp.x + sp.x) & MOD_MASK;
        int p1 = (cp.y + sp.y) & MOD_MASK;
        int p2 = (cp.z + sp.z) & MOD_MASK;
        int p3 = (cp.w + sp.w) & MOD_MASK;
        int q0 = (cm.x + sm.x) & MOD_MASK;
        int q1 = (cm.y + sm.y) & MOD_MASK;
        int q2 = (cm.z + sm.z) & MOD_MASK;
        int q3 = (cm.w + sm.w) & MOD_MASK;

        float cv0 = s_cos[p0], cv1 = s_cos[p1], cv2 = s_cos[p2], cv3 = s_cos[p3];
        float ev0 = s_exp[q0], ev1 = s_exp[q1], ev2 = s_exp[q2], ev3 = s_exp[q3];
        float dc0 = s_dcos[p0], dc1 = s_dcos[p1], dc2 = s_dcos[p2], dc3 = s_dcos[p3];
        float de0 = s_dexp[q0], de1 = s_dexp[q1], de2 = s_dexp[q2], de3 = s_dexp[q3];

        float4 po = make_float4((float)p0, (float)p1, (float)p2, (float)p3);
        float4 mo = make_float4((float)q0, (float)q1, (float)q2, (float)q3);
        float4 sg = make_float4(cv0 * ev0, cv1 * ev1, cv2 * ev2, cv3 * ev3);

        ((float4*)phase_out)[v]  = po;   // 16B-aligned (offset 0)
        ((float4*)mag_out)[v]    = mo;   // 16B-aligned (offset D, D%4==0)
        ((float4*)signal_out)[v] = sg;   // 16B-aligned (offset 2D)

        // grad arrays start at 3D+1 / 4D+1 -> 4B-aligned only: scalar stores.
        long long e = v * 4;
        gph_out[e + 0] = dc0 * ev0;
        gph_out[e + 1] = dc1 * ev1;
        gph_out[e + 2] = dc2 * ev2;
        gph_out[e + 3] = dc3 * ev3;
        gmg_out[e + 0] = cv0 * de0;
        gmg_out[e + 1] = cv1 * de1;
        gmg_out[e + 2] = cv2 * de2;
        gmg_out[e + 3] = cv3 * de3;

        acc += sg.x + sg.y + sg.z + sg.w;
    }

    // Scalar tail (D % 4 != 0), handled by block 0 only.
    long long tail = nvec * 4;
    if (blockIdx.x == 0) {
        for (long long i = tail + threadIdx.x; i < D; i += blockDim.x) {
            int p = (ctx_p[i] + self_p[i]) & MOD_MASK;
            int q = (ctx_m[i] + self_m[i]) & MOD_MASK;
            float cv = s_cos[p], ev = s_exp[q];
            float sg = cv * ev;
            phase_out[i]  = (float)p;
            mag_out[i]    = (float)q;
            signal_out[i] = sg;
            gph_out[i]    = s_dcos[p] * ev;
            gmg_out[i]    = cv * s_dexp[q];
            acc += sg;
        }
    }

    // Block reduction: wave32 shuffle tree, then LDS across the 8 waves.
    for (int off = 16; off > 0; off >>= 1)
        acc += __shfl_down(acc, off, 32);
    int lane = threadIdx.x & 31;
    int wid  = threadIdx.x >> 5;
    if (lane == 0) s_wsum[wid] = acc;
    __syncthreads();
    if (threadIdx.x == 0) {
        float t = 0.0f;
        int nw = blockDim.x >> 5;
        for (int w = 0; w < nw; ++w) t += s_wsum[w];
        partials[blockIdx.x] = t;
    }
}

// ---------------------------------------------------------------------------
// Final reduction of block partials via chained V_WMMA_F32_16X16X4_F32.
// A (16x4 f32) = 64 partials per iteration, B (4x16) = ones, C accumulates.
// A-matrix VGPR layout (ISA 7.12.2): lane (k>=2 ? 16:0)+m holds A[m][k] with
// vgpr index (k&1) -> lane loads the contiguous float2 partials[m*4+khalf*2].
// D layout: lane 0 c[0..7] = D[0..7][0], lane 16 c[0..7] = D[8..15][0];
// column 0 carries the accumulated row sums. npad must be a multiple of 64
// and the pad region zero-filled (done via hipMemsetAsync on the host side).
// One full wave => EXEC all ones as WMMA requires.
// ---------------------------------------------------------------------------
__global__ __launch_bounds__(32) void phasecell_wmma_reduce_kernel(
    const float* __restrict__ partials, float* __restrict__ strength_out,
    int npad)
{
    const int lane  = threadIdx.x;       // 0..31
    const int m     = lane & 15;
    const int khalf = lane >> 4;         // 0 or 1

    v2f b;
    b[0] = 1.0f;
    b[1] = 1.0f;
    v8f c = {};

    for (int base = 0; base < npad; base += 64) {
        const float* src = partials + base + m * 4 + khalf * 2;
        v2f a;
        a[0] = src[0];
        a[1] = src[1];
        c = __builtin_amdgcn_wmma_f32_16x16x4_f32(
            /*neg_a=*/false, a, /*neg_b=*/false, b,
            /*c_mod=*/(short)0, c, /*reuse_a=*/false, /*reuse_b=*/false);
    }

    float r = c[0] + c[1] + c[2] + c[3] + c[4] + c[5] + c[6] + c[7];
    float r0  = __shfl(r, 0, 32);   // rows 0..7  (column 0)
    float r16 = __shfl(r, 16, 32);  // rows 8..15 (column 0)
    if (lane == 0) strength_out[0] = r0 + r16;
}

extern "C" void kernel_launch(void* const* d_in, const int* in_sizes, int n_in,
                              void* d_out, int out_size, void* d_ws, size_t ws_size,
                              hipStream_t stream) {
    const int*   ctx_p  = (const int*)d_in[0];
    const int*   ctx_m  = (const int*)d_in[1];
    const int*   self_p = (const int*)d_in[2];
    const int*   self_m = (const int*)d_in[3];
    const float* cos_t  = (const float*)d_in[4];
    const float* exp_t  = (const float*)d_in[5];
    const float* dcos_t = (const float*)d_in[6];
    const float* dexp_t = (const float*)d_in[7];

    const long long D = (long long)in_sizes[0];
    float* out        = (float*)d_out;
    float* phase_out  = out;                 // [D]
    float* mag_out    = out + D;             // [D]
    float* signal_out = out + 2 * D;         // [D]
    float* strength   = out + 3 * D;         // [1]
    float* gph_out    = out + 3 * D + 1;     // [D]
    float* gmg_out    = out + 4 * D + 1;     // [D]

    float* partials = (float*)d_ws;

    const long long nvec = D / 4;
    const int block = 256;
    long long need = (nvec + block - 1) / block;
    if (need < 1) need = 1;
    long long cap = 8192;                    // 32 KB of partials
    long long wscap = (long long)(ws_size / sizeof(float));
    if (wscap / 64 * 64 < cap) cap = wscap / 64 * 64;
    if (cap < 64) cap = 64;
    const int grid = (int)(need < cap ? need : cap);
    const int npad = ((grid + 63) / 64) * 64;

    // Zero (grid..npad) pad region so the WMMA reduction needs no predication.
    hipMemsetAsync(partials, 0, (size_t)npad * sizeof(float), stream);

    phasecell_main_kernel<<<grid, block, 0, stream>>>(
        ctx_p, ctx_m, self_p, self_m,
        cos_t, exp_t, dcos_t, dexp_t,
        phase_out, mag_out, signal_out, gph_out, gmg_out,
        partials, nvec, D);

    phasecell_wmma_reduce_kernel<<<1, 32, 0, stream>>>(partials, strength, npad);
}